// GCKANLayer_19559281066593
// MI455X (gfx1250) — compile-verified
//
#include <hip/hip_runtime.h>
#include <hip/hip_bf16.h>

// ---------------------------------------------------------------------------
// GCKAN layer for MI455X (gfx1250, wave32, WMMA + async global->LDS).
//  K1: h   = x @ W_node^T + b_node                      (WMMA f32 16x16x4)
//  K2: e   = edge_attr @ W_edge^T + b_edge; msg = e*h[row]; atomicAdd->agg
//      (A tile async-staged to LDS: global_load_async_to_lds_b128)
//  K3: BT  = [base_weight | spline_weight*scaler] permuted to chunked K order
//  K4: out = [silu(agg) | bspline(agg)] @ BT^T          (LDS-staged WMMA)
// ---------------------------------------------------------------------------

typedef __attribute__((ext_vector_type(2))) float v2f;
typedef __attribute__((ext_vector_type(8))) float v8f;

#define N_NODES   10000
#define N_EDGES   300000
#define IN_FEAT   256
#define OUT_FEAT  256
#define EDGE_DIM  384
#define NCOEF     7          // grid_size + spline_order = 4 + 3
#define KAN_K     2048       // 256 silu cols + 256*7 spline cols
#define CHUNK_CH  64         // channels per LDS chunk in K4
#define CHUNK_K   512        // 64 + 64*7 columns per chunk
#define EM_TILE   32         // edges per block in K2

// D = A(16x4) * B(4x16) + C, fp32. 8-arg form:
// (neg_a, A, neg_b, B, c_mod, C, reuse_a, reuse_b)
__device__ __forceinline__ v8f wmma_f32(v2f a, v2f b, v8f c) {
    return __builtin_amdgcn_wmma_f32_16x16x4_f32(
        false, a, false, b, (short)0, c, false, false);
}

// Async DMA of 16B/lane from global memory into LDS (ASYNCcnt-tracked).
// lds_off: wave-relative LDS byte offset (VDST VGPR); gptr: 64-bit address.
__device__ __forceinline__ void async_load_b128(unsigned int lds_off,
                                                const void* gptr) {
    asm volatile("global_load_async_to_lds_b128 %0, %1, off"
                 :: "v"(lds_off), "v"(gptr) : "memory");
}
__device__ __forceinline__ void wait_async0() {
    asm volatile("s_wait_asynccnt 0x0" ::: "memory");
}

__global__ void zero_f32_kernel(float* __restrict__ p, int n) {
    int i = blockIdx.x * blockDim.x + threadIdx.x;
    if (i < n) p[i] = 0.0f;
}

// ---------------------------------------------------------------------------
// K1: h[m][n] = sum_k x[m][k] * W_node[n][k] + b_node[n]
// Block: 128 threads (4 waves). Block tile M=16, each wave N=64 (4 wmma tiles).
// ---------------------------------------------------------------------------
__global__ __launch_bounds__(128)
void node_linear_kernel(const float* __restrict__ x,
                        const float* __restrict__ W,
                        const float* __restrict__ bias,
                        float* __restrict__ h) {
    const int m_base = blockIdx.x * 16;
    const int lane   = threadIdx.x & 31;
    const int wave   = threadIdx.x >> 5;
    const int lq     = lane & 15;      // A: row M / B: col N / C: col N
    const int half   = lane >> 4;      // K-pair select (A/B), +8 rows (C)
    const int kh     = half * 2;

    v8f acc[4] = {};
    const float* arow = x + (size_t)(m_base + lq) * IN_FEAT;
    for (int k = 0; k < IN_FEAT; k += 4) {
        __builtin_prefetch(arow + k + 32, 0, 0);
        v2f a = *(const v2f*)(arow + k + kh);
#pragma unroll
        for (int t = 0; t < 4; ++t) {
            const int n = wave * 64 + t * 16 + lq;
            v2f b = *(const v2f*)(W + (size_t)n * IN_FEAT + k + kh);
            acc[t] = wmma_f32(a, b, acc[t]);
        }
    }
#pragma unroll
    for (int t = 0; t < 4; ++t) {
        const int n  = wave * 64 + t * 16 + lq;
        const float bv = bias[n];
#pragma unroll
        for (int r = 0; r < 8; ++r) {
            const int m = m_base + r + half * 8;
            h[(size_t)m * OUT_FEAT + n] = acc[t][r] + bv;
        }
    }
}

// ---------------------------------------------------------------------------
// K2: fused edge linear + message + scatter. M-tile = 32 edges per block.
//  A tile (32x384 fp32, contiguous 48KB) async-staged into LDS, then
//  each wave computes two 16-row WMMA sub-tiles sharing every B fragment.
//  msg = (e + b_edge) * h[row];  atomicAdd(agg[col][n], msg)  (agg in L2)
// ---------------------------------------------------------------------------
__global__ __launch_bounds__(128)
void edge_msg_kernel(const float* __restrict__ ea,
                     const float* __restrict__ W,
                     const float* __restrict__ bias,
                     const int*   __restrict__ eidx,
                     const float* __restrict__ h,
                     float* __restrict__ agg) {
    __shared__ float As[EM_TILE * EDGE_DIM];   // 48 KB
    __shared__ int s_row[EM_TILE];
    __shared__ int s_col[EM_TILE];
    const int e_base = blockIdx.x * EM_TILE;

    if (threadIdx.x < EM_TILE)
        s_row[threadIdx.x] = eidx[e_base + threadIdx.x];
    else if (threadIdx.x < 2 * EM_TILE)
        s_col[threadIdx.x - EM_TILE] = eidx[N_EDGES + e_base + threadIdx.x - EM_TILE];

    // Async DMA: 32 contiguous edge_attr rows -> LDS, 16B per lane per op.
    {
        const unsigned int lds0 = (unsigned int)(uintptr_t)(&As[0]);
        const float* g0 = ea + (size_t)e_base * EDGE_DIM;
        for (int c = threadIdx.x; c < (EM_TILE * EDGE_DIM) / 4; c += 128)
            async_load_b128(lds0 + (unsigned int)c * 16u, g0 + (size_t)c * 4);
        wait_async0();
    }
    __syncthreads();

    const int lane = threadIdx.x & 31;
    const int wave = threadIdx.x >> 5;
    const int lq   = lane & 15;
    const int half = lane >> 4;
    const int kh   = half * 2;

    v8f acc[2][4] = {};
    const float* a0row = As + lq * EDGE_DIM;
    const float* a1row = As + (lq + 16) * EDGE_DIM;
    for (int k = 0; k < EDGE_DIM; k += 4) {
        v2f a0 = *(const v2f*)(a0row + k + kh);
        v2f a1 = *(const v2f*)(a1row + k + kh);
#pragma unroll
        for (int t = 0; t < 4; ++t) {
            const int n = wave * 64 + t * 16 + lq;
            v2f b = *(const v2f*)(W + (size_t)n * EDGE_DIM + k + kh);
            acc[0][t] = wmma_f32(a0, b, acc[0][t]);
            acc[1][t] = wmma_f32(a1, b, acc[1][t]);
        }
    }
#pragma unroll
    for (int ms = 0; ms < 2; ++ms) {
#pragma unroll
        for (int t = 0; t < 4; ++t) {
            const int n  = wave * 64 + t * 16 + lq;
            const float bv = bias[n];
#pragma unroll
            for (int r = 0; r < 8; ++r) {
                const int m    = ms * 16 + r + half * 8;
                const float ev = acc[ms][t][r] + bv;
                const float msg = ev * h[(size_t)s_row[m] * OUT_FEAT + n];
                atomicAdd(&agg[(size_t)s_col[m] * OUT_FEAT + n], msg);
            }
        }
    }
}

// ---------------------------------------------------------------------------
// K3: build combined KAN weight matrix BT[o][k], k in chunk-permuted order:
//   chunk = k/512, local = k%512
//   local <  64 : base column  i = chunk*64 + local        -> base_weight[o][i]
//   local >= 64 : spline col   i = chunk*64 + (local-64)/7, c = (local-64)%7
//                 -> spline_weight[o][i][c] * spline_scaler[o][i]
// ---------------------------------------------------------------------------
__global__ void build_bmat_kernel(const float* __restrict__ base_w,
                                  const float* __restrict__ spline_w,
                                  const float* __restrict__ scaler,
                                  float* __restrict__ BT) {
    int idx = blockIdx.x * blockDim.x + threadIdx.x;   // o*2048 + k
    if (idx >= OUT_FEAT * KAN_K) return;
    const int o     = idx >> 11;
    const int k     = idx & (KAN_K - 1);
    const int chunk = k >> 9;
    const int local = k & (CHUNK_K - 1);
    float v;
    if (local < CHUNK_CH) {
        const int i = chunk * CHUNK_CH + local;
        v = base_w[o * OUT_FEAT + i];
    } else {
        const int sp = local - CHUNK_CH;
        const int i  = chunk * CHUNK_CH + sp / NCOEF;
        const int c  = sp - NCOEF * (sp / NCOEF);
        v = spline_w[(size_t)(o * OUT_FEAT + i) * NCOEF + c] * scaler[o * OUT_FEAT + i];
    }
    BT[idx] = v;
}

// ---------------------------------------------------------------------------
// K4: KANLinear. Per block: 16 nodes. 4 chunks of 64 channels:
//  stage LDS A[16][512] = [silu(a) | 7 cubic B-spline bases per channel],
//  then WMMA-GEMM A(16x512) * BT_chunk^T(512x256) accumulating into C.
// Uniform knots t_j = -2.5 + 0.5j (grid_size=4, k=3, range [-1,1]).
// ---------------------------------------------------------------------------
__global__ __launch_bounds__(128)
void kan_kernel(const float* __restrict__ agg,
                const float* __restrict__ BT,
                float* __restrict__ out) {
    __shared__ float As[16 * CHUNK_K];   // 32 KB
    const int m_base = blockIdx.x * 16;
    const int lane   = threadIdx.x & 31;
    const int wave   = threadIdx.x >> 5;
    const int lq     = lane & 15;
    const int half   = lane >> 4;
    const int kh     = half * 2;

    v8f acc[4] = {};
    for (int ch = 0; ch < 4; ++ch) {
        __syncthreads();   // previous GEMM done reading As
        // ---- stage 1: 16 nodes x 64 channels -> silu + spline basis in LDS
        for (int p = threadIdx.x; p < 16 * CHUNK_CH; p += 128) {
            const int m = p >> 6;
            const int i = p & (CHUNK_CH - 1);
            const float a = agg[(size_t)(m_base + m) * OUT_FEAT + ch * CHUNK_CH + i];
            As[m * CHUNK_K + i] = a / (1.0f + __expf(-a));   // silu
            // Cox-de Boor, degree 0..3, uniform knots, h = 0.5
            float b0[10];
#pragma unroll
            for (int j = 0; j < 10; ++j) {
                const float tj = -2.5f + 0.5f * (float)j;
                b0[j] = (a >= tj && a < tj + 0.5f) ? 1.0f : 0.0f;
            }
#pragma unroll
            for (int d = 1; d <= 3; ++d) {
                const float inv = 1.0f / (0.5f * (float)d);
                for (int j = 0; j + d < 10; ++j) {
                    const float tj = -2.5f + 0.5f * (float)j;
                    b0[j] = (a - tj) * inv * b0[j]
                          + (tj + 0.5f * (float)(d + 1) - a) * inv * b0[j + 1];
                }
            }
#pragma unroll
            for (int c = 0; c < NCOEF; ++c)
                As[m * CHUNK_K + CHUNK_CH + i * NCOEF + c] = b0[c];
        }
        __syncthreads();
        // ---- stage 2: WMMA GEMM over this chunk's 512 K-columns
        const float* arow = As + lq * CHUNK_K;
        for (int k = 0; k < CHUNK_K; k += 4) {
            v2f a = *(const v2f*)(arow + k + kh);
#pragma unroll
            for (int t = 0; t < 4; ++t) {
                const int n = wave * 64 + t * 16 + lq;
                v2f b = *(const v2f*)(BT + (size_t)n * KAN_K + ch * CHUNK_K + k + kh);
                acc[t] = wmma_f32(a, b, acc[t]);
            }
        }
    }
#pragma unroll
    for (int t = 0; t < 4; ++t) {
        const int n = wave * 64 + t * 16 + lq;
#pragma unroll
        for (int r = 0; r < 8; ++r) {
            const int m = m_base + r + half * 8;
            out[(size_t)m * OUT_FEAT + n] = acc[t][r];
        }
    }
}

// ---------------------------------------------------------------------------
extern "C" void kernel_launch(void* const* d_in, const int* in_sizes, int n_in,
                              void* d_out, int out_size, void* d_ws, size_t ws_size,
                              hipStream_t stream) {
    const float* x          = (const float*)d_in[0];
    const float* edge_attr  = (const float*)d_in[1];
    const int*   edge_index = (const int*)  d_in[2];
    const float* W_edge     = (const float*)d_in[3];
    const float* b_edge     = (const float*)d_in[4];
    const float* W_node     = (const float*)d_in[5];
    const float* b_node     = (const float*)d_in[6];
    const float* base_w     = (const float*)d_in[7];
    const float* spline_w   = (const float*)d_in[8];
    const float* scaler     = (const float*)d_in[9];
    float* out = (float*)d_out;

    // workspace layout (all fully overwritten every call)
    float* h   = (float*)d_ws;                           // 10000*256
    float* agg = h   + (size_t)N_NODES * OUT_FEAT;       // 10000*256
    float* BT  = agg + (size_t)N_NODES * OUT_FEAT;       // 256*2048

    zero_f32_kernel<<<(N_NODES * OUT_FEAT + 255) / 256, 256, 0, stream>>>(
        agg, N_NODES * OUT_FEAT);
    build_bmat_kernel<<<(OUT_FEAT * KAN_K + 255) / 256, 256, 0, stream>>>(
        base_w, spline_w, scaler, BT);
    node_linear_kernel<<<N_NODES / 16, 128, 0, stream>>>(x, W_node, b_node, h);
    edge_msg_kernel<<<N_EDGES / EM_TILE, 128, 0, stream>>>(
        edge_attr, W_edge, b_edge, edge_index, h, agg);
    kan_kernel<<<N_NODES / 16, 128, 0, stream>>>(agg, BT, out);
}